// EMAQuantizer_54786602828224
// MI455X (gfx1250) — compile-verified
//
#include <hip/hip_runtime.h>
#include <hip/hip_bf16.h>

typedef __attribute__((ext_vector_type(2))) float v2f;
typedef __attribute__((ext_vector_type(8))) float v8f;

#define N_EMBED   8192
#define EMBED_DIM 64
#define N_ROWS    16384     // 16*32*32
#define HW        1024      // 32*32
#define TOT_ELEMS 1048576   // 16*64*32*32

// ---------------------------------------------------------------------------
// Kernel 0: wsq[k] = sum_d weight[k][d]^2
// ---------------------------------------------------------------------------
__global__ __launch_bounds__(256) void wsq_kernel(const float* __restrict__ w,
                                                  float* __restrict__ wsq) {
    int k = blockIdx.x * 256 + threadIdx.x;
    if (k < N_EMBED) {
        const float4* wr = (const float4*)(w + k * EMBED_DIM);
        float s = 0.0f;
#pragma unroll
        for (int i = 0; i < 16; ++i) {
            float4 v = wr[i];
            s += v.x * v.x + v.y * v.y + v.z * v.z + v.w * v.w;
        }
        wsq[k] = s;
    }
}

// ---------------------------------------------------------------------------
// Kernel 1: fused distance GEMM (fp32 WMMA 16x16x4) + running argmin.
// One wave per 16-row M-tile of z_flat. score = wsq[k] - 2*z.w  (||z||^2 const)
// A layout (16x4 f32): lane L -> row M=L%16; half=L/16 selects K {0,1}/{2,3}
// B layout (4x16 f32): lane L -> col N=L%16; half selects same K pairs
// C/D layout: VGPR r, lane L -> row r+8*(L>>4), col L&15
// ---------------------------------------------------------------------------
__global__ __launch_bounds__(256) void argmin_kernel(const float* __restrict__ z,
                                                     const float* __restrict__ weight,
                                                     const float* __restrict__ wsq,
                                                     int* __restrict__ idx_out) {
    const int lane    = threadIdx.x & 31;
    const int mtile   = blockIdx.x * 8 + (threadIdx.x >> 5);   // 1024 M-tiles
    const int colLane = lane & 15;
    const int half    = lane >> 4;

    // z_flat[row][d] = z[b*65536 + d*1024 + pos], row = b*1024 + pos
    const int row = mtile * 16 + colLane;
    const int b   = row >> 10;
    const int pos = row & 1023;
    const float* zbase = z + b * (EMBED_DIM * HW) + pos;

    // Load A once for full K=64, pre-scaled by -2.
    v2f a[16];
#pragma unroll
    for (int s = 0; s < 16; ++s) {
        int d0 = 4 * s + 2 * half;
        a[s].x = -2.0f * zbase[d0 * HW];
        a[s].y = -2.0f * zbase[(d0 + 1) * HW];
    }

    float minv[8];
    int   mini[8];
#pragma unroll
    for (int r = 0; r < 8; ++r) { minv[r] = 3.4e38f; mini[r] = 0; }

    for (int kt = 0; kt < N_EMBED / 16; ++kt) {
        const int code = kt * 16 + colLane;
        const float* wrow = weight + code * EMBED_DIM + 2 * half;

        v2f bm[16];
#pragma unroll
        for (int s = 0; s < 16; ++s)
            bm[s] = *(const v2f*)(wrow + 4 * s);

        const float c0 = wsq[code];
        v8f acc = { c0, c0, c0, c0, c0, c0, c0, c0 };

#pragma unroll
        for (int s = 0; s < 16; ++s)
            acc = __builtin_amdgcn_wmma_f32_16x16x4_f32(
                false, a[s], false, bm[s], (short)0, acc, false, false);

#pragma unroll
        for (int r = 0; r < 8; ++r) {
            float v = acc[r];
            if (v < minv[r]) { minv[r] = v; mini[r] = code; }  // strict < keeps first
        }
    }

    // Reduce (min,argmin) across the 16 lanes sharing each row (xor 1,2,4,8
    // never crosses the half boundary). Tie-break toward smaller index.
#pragma unroll
    for (int r = 0; r < 8; ++r) {
        float v = minv[r];
        int   i = mini[r];
#pragma unroll
        for (int m = 1; m <= 8; m <<= 1) {
            float ov = __shfl_xor(v, m, 32);
            int   oi = __shfl_xor(i, m, 32);
            if (ov < v || (ov == v && oi < i)) { v = ov; i = oi; }
        }
        minv[r] = v; mini[r] = i;
    }

    if (colLane == 0) {
#pragma unroll
        for (int r = 0; r < 8; ++r)
            idx_out[mtile * 16 + half * 8 + r] = mini[r];
    }
}

// ---------------------------------------------------------------------------
// Kernel 2: z_q gather (already in b,c,h,w output layout) + loss partials.
// o = ((b*64+c)*32+h)*32+w ; z has the same layout; n = b*1024 + (h*32+w)
// ---------------------------------------------------------------------------
__global__ __launch_bounds__(256) void gather_loss_kernel(const float* __restrict__ z,
                                                          const float* __restrict__ weight,
                                                          const int* __restrict__ idx,
                                                          float* __restrict__ zq_out,
                                                          float* __restrict__ loss_part) {
    __shared__ float sdata[256];
    const int o   = blockIdx.x * 256 + threadIdx.x;
    const int pos = o & 1023;
    const int c   = (o >> 10) & 63;
    const int b   = o >> 16;
    const int n   = (b << 10) + pos;

    const float q  = weight[idx[n] * EMBED_DIM + c];
    const float zv = z[o];
    zq_out[o] = q;                      // straight-through forward value == z_q
    const float d = q - zv;
    sdata[threadIdx.x] = d * d;
    __syncthreads();
    for (int st = 128; st > 0; st >>= 1) {
        if (threadIdx.x < st) sdata[threadIdx.x] += sdata[threadIdx.x + st];
        __syncthreads();
    }
    if (threadIdx.x == 0) loss_part[blockIdx.x] = sdata[0];
}

// ---------------------------------------------------------------------------
// Kernel 3: deterministic segment sums + EMA. One 64-thread block per code k;
// thread d accumulates dim d over n in ascending order (fixed fp order).
// ---------------------------------------------------------------------------
__global__ __launch_bounds__(64) void segsum_ema_kernel(const float* __restrict__ z,
                                                        const int* __restrict__ idx,
                                                        const float* __restrict__ cluster_size,
                                                        const float* __restrict__ embed_avg,
                                                        float* __restrict__ out_cs,
                                                        float* __restrict__ out_ea,
                                                        float* __restrict__ cs_copy) {
    const int k = blockIdx.x;
    const int d = threadIdx.x;
    float acc = 0.0f;
    int   cnt = 0;
    const int4* idx4 = (const int4*)idx;
    for (int n4 = 0; n4 < N_ROWS / 4; ++n4) {
        int4 m = idx4[n4];
        int  n = n4 * 4;
        if (m.x == k) { int bb = n >> 10;       acc += z[bb * 65536 + d * HW + (n & 1023)];       ++cnt; }
        if (m.y == k) { int bb = (n + 1) >> 10; acc += z[bb * 65536 + d * HW + ((n + 1) & 1023)]; ++cnt; }
        if (m.z == k) { int bb = (n + 2) >> 10; acc += z[bb * 65536 + d * HW + ((n + 2) & 1023)]; ++cnt; }
        if (m.w == k) { int bb = (n + 3) >> 10; acc += z[bb * 65536 + d * HW + ((n + 3) & 1023)]; ++cnt; }
    }
    out_ea[k * EMBED_DIM + d] = embed_avg[k * EMBED_DIM + d] * 0.99f + 0.01f * acc;
    if (d == 0) {
        float ncs = cluster_size[k] * 0.99f + 0.01f * (float)cnt;
        out_cs[k]  = ncs;
        cs_copy[k] = ncs;
    }
}

// ---------------------------------------------------------------------------
// Kernel 4: finalize loss and n = sum(new_cluster_size). Single block.
// ---------------------------------------------------------------------------
__global__ __launch_bounds__(1024) void finalize_kernel(const float* __restrict__ loss_part,
                                                        const float* __restrict__ cs_copy,
                                                        float* __restrict__ loss_out,
                                                        float* __restrict__ n_out) {
    __shared__ float s1[1024];
    __shared__ float s2[1024];
    float a = 0.0f, b = 0.0f;
    for (int i = threadIdx.x; i < 4096; i += 1024) a += loss_part[i];
    for (int i = threadIdx.x; i < N_EMBED; i += 1024) b += cs_copy[i];
    s1[threadIdx.x] = a;
    s2[threadIdx.x] = b;
    __syncthreads();
    for (int st = 512; st > 0; st >>= 1) {
        if (threadIdx.x < st) {
            s1[threadIdx.x] += s1[threadIdx.x + st];
            s2[threadIdx.x] += s2[threadIdx.x + st];
        }
        __syncthreads();
    }
    if (threadIdx.x == 0) {
        *loss_out = 0.25f * s1[0] / (float)TOT_ELEMS;   // BETA * mean
        *n_out    = s2[0];
    }
}

// ---------------------------------------------------------------------------
// Kernel 5: new_weight = new_embed_avg / smoothed(new_cluster_size, n)
// ---------------------------------------------------------------------------
__global__ __launch_bounds__(256) void weight_kernel(const float* __restrict__ ea,
                                                     const float* __restrict__ cs,
                                                     const float* __restrict__ n_ptr,
                                                     float* __restrict__ w_out) {
    const int t = blockIdx.x * 256 + threadIdx.x;   // 524288 elements
    const int k = t >> 6;
    const float n  = *n_ptr;
    const float sm = (cs[k] + 1e-5f) / (n + (float)N_EMBED * 1e-5f) * n;
    w_out[t] = ea[t] / sm;
}

// ---------------------------------------------------------------------------
extern "C" void kernel_launch(void* const* d_in, const int* in_sizes, int n_in,
                              void* d_out, int out_size, void* d_ws, size_t ws_size,
                              hipStream_t stream) {
    const float* z            = (const float*)d_in[0];   // [16,64,32,32]
    const float* weight       = (const float*)d_in[1];   // [8192,64]
    const float* cluster_size = (const float*)d_in[2];   // [8192]
    const float* embed_avg    = (const float*)d_in[3];   // [8192,64]

    // d_out: z_q | loss | new_weight | new_cluster_size | new_embed_avg
    float* out    = (float*)d_out;
    float* zq     = out;
    float* loss   = out + TOT_ELEMS;                   // 1048576
    float* w_new  = out + TOT_ELEMS + 1;               // 1048577
    float* cs_new = out + TOT_ELEMS + 1 + N_EMBED * EMBED_DIM;          // 1572865
    float* ea_new = out + TOT_ELEMS + 1 + N_EMBED * EMBED_DIM + N_EMBED;// 1581057

    // d_ws: wsq[8192] | idx[16384] (int) | loss_part[4096] | cs_copy[8192] | n[1]
    float* ws        = (float*)d_ws;
    float* wsq       = ws;
    int*   idx       = (int*)(ws + N_EMBED);
    float* loss_part = ws + N_EMBED + N_ROWS;
    float* cs_copy   = loss_part + 4096;
    float* n_val     = cs_copy + N_EMBED;

    wsq_kernel<<<N_EMBED / 256, 256, 0, stream>>>(weight, wsq);
    argmin_kernel<<<(N_ROWS / 16) / 8, 256, 0, stream>>>(z, weight, wsq, idx);
    gather_loss_kernel<<<TOT_ELEMS / 256, 256, 0, stream>>>(z, weight, idx, zq, loss_part);
    segsum_ema_kernel<<<N_EMBED, 64, 0, stream>>>(z, idx, cluster_size, embed_avg,
                                                  cs_new, ea_new, cs_copy);
    finalize_kernel<<<1, 1024, 0, stream>>>(loss_part, cs_copy, loss, n_val);
    weight_kernel<<<(N_EMBED * EMBED_DIM) / 256, 256, 0, stream>>>(ea_new, cs_new, n_val, w_new);
}